// CRF_16432544875067
// MI455X (gfx1250) — compile-verified
//
#include <hip/hip_runtime.h>
#include <math.h>

// CRF loss: mean_b( log_partition(b) - path_score(b) )
// B=128, S=2048, T=96 tags (+START=96, +END=97 -> 98x98 transitions)

#define T_TAGS    96
#define T_DIM     98          // T + 2
#define S_LEN     2048
#define B_TOT     128
#define START_TAG 96
#define END_TAG   97
#define NEGV      (-10000.0f)
#define ALN       100         // padded LDS row stride (floats); 400B, 16B-aligned

typedef __attribute__((ext_vector_type(2))) float v2f;
typedef __attribute__((ext_vector_type(4))) float v4f;
typedef __attribute__((ext_vector_type(8))) float v8f;

// ---------------------------------------------------------------------------
// gfx1250 async global->LDS copy (ASYNCcnt path) + waits
// LDS address operand = low 32 bits of the generic pointer (LDS aperture).
// ---------------------------------------------------------------------------
__device__ __forceinline__ void async_load_b128(const float* g, float* l) {
    unsigned lds = (unsigned)(unsigned long long)l;
    asm volatile("global_load_async_to_lds_b128 %0, %1, off"
                 :: "v"(lds), "v"(g) : "memory");
}
__device__ __forceinline__ void wait_async0() {
    asm volatile("s_wait_asynccnt 0" ::: "memory");
}
__device__ __forceinline__ void wait_async2() {
    asm volatile("s_wait_asynccnt 2" ::: "memory");   // newest 2 may remain
}

// ---------------------------------------------------------------------------
// Kernel 0: expT[i][j] = exp(transitions[i][j]) for i,j < 96 (row-major 96x96)
// ---------------------------------------------------------------------------
__global__ void crf_exptrans(const float* __restrict__ trans,
                             float* __restrict__ expT) {
    int idx = blockIdx.x * blockDim.x + threadIdx.x;
    if (idx < T_TAGS * T_TAGS) {
        int i = idx / T_TAGS;
        int j = idx - i * T_TAGS;
        expT[idx] = __expf(trans[i * T_DIM + j]);
    }
}

// ---------------------------------------------------------------------------
// Kernel 1: numerator path score per batch row
// ---------------------------------------------------------------------------
__global__ void __launch_bounds__(256)
crf_score(const float* __restrict__ emissions, const int* __restrict__ tags,
          const float* __restrict__ mask, const float* __restrict__ trans,
          float* __restrict__ log_num) {
    __shared__ float sScore[256];
    __shared__ float sMaskR[256];
    const int b = blockIdx.x;
    const int tid = threadIdx.x;

    float sc = 0.0f, ms = 0.0f;
    for (int t = tid; t < S_LEN; t += 256) {
        int tag  = tags[(size_t)b * S_LEN + t];
        int prev = (t == 0) ? START_TAG : tags[(size_t)b * S_LEN + t - 1];
        float m  = mask[(size_t)b * S_LEN + t];
        float e  = emissions[((size_t)b * S_LEN + t) * T_TAGS + tag];
        sc += (e + trans[prev * T_DIM + tag]) * m;
        ms += m;
    }
    sScore[tid] = sc; sMaskR[tid] = ms;
    __syncthreads();
    for (int s = 128; s > 0; s >>= 1) {
        if (tid < s) { sScore[tid] += sScore[tid + s]; sMaskR[tid] += sMaskR[tid + s]; }
        __syncthreads();
    }
    if (tid == 0) {
        int last_idx = (int)(sMaskR[0] + 0.5f);
        int last_tag = (last_idx == 0) ? START_TAG
                                       : tags[(size_t)b * S_LEN + last_idx - 1];
        log_num[b] = sScore[0] + trans[last_tag * T_DIM + END_TAG];
    }
}

// ---------------------------------------------------------------------------
// Kernel 2: forward algorithm. fp32 WMMA GEMM per step, alpha in registers,
// one-step-stale row scaling, triple-buffered async emissions, ONE barrier
// per step. Block = 16 batch rows, 6 waves (192 thr); wave w owns N-cols
// [16w, 16w+16).
// ---------------------------------------------------------------------------
__global__ void __launch_bounds__(192)
crf_partition(const float* __restrict__ emissions,
              const float* __restrict__ mask,
              const float* __restrict__ trans,
              const float* __restrict__ expT,
              float* __restrict__ log_den) {
    __shared__ float sExpA[2][16][ALN];                      // exp(alpha - Msc)
    __shared__ float sEmitBuf[3][16][ALN];                   // async emissions
    __shared__ __attribute__((aligned(16))) float sPart[2][16][8];  // row maxes
    __shared__ float sMaskC[2][16][32];                      // mask chunks

    const int tid  = threadIdx.x;
    const int lane = tid & 31;
    const int wave = tid >> 5;                     // 0..5 : N-tile
    const int n0   = wave * 16;
    const int b0   = blockIdx.x * 16;

    const int row = tid / 12;                      // staging row 0..15
    const int j0  = (tid % 12) * 8;                // 8 floats -> two B128

    const int lh = lane >> 4;                      // K-pair / M-half select
    const int ln = lane & 15;                      // A-row / B,D-column

    // ---- preload B fragments: expT 96x16 slice, 24 K-steps of 4 ----
    v2f bfrag[24];
#pragma unroll
    for (int kk = 0; kk < 24; ++kk) {
        int k = kk * 4 + lh * 2;
        bfrag[kk].x = expT[(k    ) * T_TAGS + n0 + ln];
        bfrag[kk].y = expT[(k + 1) * T_TAGS + n0 + ln];
    }

    // ---- mask chunk 0 (steps 0..31) ----
    if (tid < 128) {
        int rr = tid >> 3, ri = (tid & 7) * 4;
        v4f mv = *(const v4f*)(mask + (size_t)(b0 + rr) * S_LEN + ri);
        *(v4f*)&sMaskC[0][rr][ri] = mv;
    }
    // ---- async emissions for t = 1 (buf 1) and t = 2 (buf 2) ----
    {
        const float* g1 = emissions + ((size_t)(b0 + row) * S_LEN + 1) * T_TAGS + j0;
        async_load_b128(g1,     &sEmitBuf[1][row][j0]);
        async_load_b128(g1 + 4, &sEmitBuf[1][row][j0 + 4]);
        const float* g2 = g1 + T_TAGS;
        async_load_b128(g2,     &sEmitBuf[2][row][j0]);
        async_load_b128(g2 + 4, &sEmitBuf[2][row][j0 + 4]);
    }

    // ---- alpha at t=0 (registers): m0*(trans[START,j]+emit0) + (1-m0)*NEG ----
    float alphaReg[8], MscReg[8];
    {
        float trS = trans[START_TAG * T_DIM + n0 + ln];
#pragma unroll
        for (int r = 0; r < 8; ++r) {
            int b = b0 + lh * 8 + r;
            float e  = emissions[(size_t)b * S_LEN * T_TAGS + n0 + ln];
            float m0 = mask[(size_t)b * S_LEN];
            alphaReg[r] = (trS + e) * m0 + NEGV * (1.0f - m0);
        }
    }
    // row maxes of alpha_0 -> sPart[1] (read by step 1)
#pragma unroll
    for (int r = 0; r < 8; ++r) {
        float v = alphaReg[r];
#pragma unroll
        for (int o = 1; o < 16; o <<= 1) v = fmaxf(v, __shfl_xor(v, o, 16));
        if (ln == 0) sPart[1][lh * 8 + r][wave] = v;
    }
    __syncthreads();
    // exp(alpha_0 - rowmax(alpha_0)) -> sExpA[1] (read by step 1)
#pragma unroll
    for (int r = 0; r < 8; ++r) {
        int m = lh * 8 + r;
        v4f lo = *(const v4f*)&sPart[1][m][0];
        v4f hi = *(const v4f*)&sPart[1][m][4];
        float v = fmaxf(fmaxf(fmaxf(lo.x, lo.y), fmaxf(lo.z, lo.w)),
                        fmaxf(hi.x, hi.y));
        MscReg[r] = v;
        sExpA[1][m][n0 + ln] = __expf(alphaReg[r] - v);
    }
    wait_async2();                  // t=1's emissions done (t=2's may remain)
    __syncthreads();

    // ---- sequential forward steps: ONE barrier each ----
    for (int t = 1; t < S_LEN; ++t) {
        const int p  = t & 1;       // sExpA read buffer
        const int q  = t & 1;       // sPart read buffer
        const int pb = t % 3;       // emissions buffer for this step

        // issue async emissions for t+2
        if (t + 2 < S_LEN) {
            const float* g = emissions +
                ((size_t)(b0 + row) * S_LEN + (t + 2)) * T_TAGS + j0;
            async_load_b128(g,     &sEmitBuf[(t + 2) % 3][row][j0]);
            async_load_b128(g + 4, &sEmitBuf[(t + 2) % 3][row][j0 + 4]);
            __builtin_prefetch(g + 2 * T_TAGS, 0, 3);       // warm for t+4
        }

        // WMMA: P(16x16) = expA(16x96) x expT(96x16), two interleaved chains
        v8f acc0 = {}, acc1 = {};
#pragma unroll
        for (int kk = 0; kk < 24; kk += 2) {
            int ka = kk * 4 + lh * 2;
            v2f a0; a0.x = sExpA[p][ln][ka]; a0.y = sExpA[p][ln][ka + 1];
            acc0 = __builtin_amdgcn_wmma_f32_16x16x4_f32(
                false, a0, false, bfrag[kk], (short)0, acc0, false, false);
            int kb = (kk + 1) * 4 + lh * 2;
            v2f a1; a1.x = sExpA[p][ln][kb]; a1.y = sExpA[p][ln][kb + 1];
            acc1 = __builtin_amdgcn_wmma_f32_16x16x4_f32(
                false, a1, false, bfrag[kk + 1], (short)0, acc1, false, false);
        }

        // alpha' = log(P) + Msc + emit ; masked blend against register alpha
        const int mslot = t & 31, mbuf = (t >> 5) & 1;
#pragma unroll
        for (int r = 0; r < 8; ++r) {
            int m = lh * 8 + r;
            float pp = fmaxf(acc0[r] + acc1[r], 1e-37f);
            float at = __logf(pp) + MscReg[r] + sEmitBuf[pb][m][n0 + ln];
            float mm = sMaskC[mbuf][m][mslot];
            alphaReg[r] = at * mm + alphaReg[r] * (1.0f - mm);
        }

        // scale for next step = rowmax(alpha_{t-1}) from sPart[q] (stale by 1);
        // write exp(alpha_t - Msc) into the other sExpA buffer
#pragma unroll
        for (int r = 0; r < 8; ++r) {
            int m = lh * 8 + r;
            v4f lo = *(const v4f*)&sPart[q][m][0];
            v4f hi = *(const v4f*)&sPart[q][m][4];
            float v = fmaxf(fmaxf(fmaxf(lo.x, lo.y), fmaxf(lo.z, lo.w)),
                            fmaxf(hi.x, hi.y));
            MscReg[r] = v;
            sExpA[p ^ 1][m][n0 + ln] = __expf(alphaReg[r] - v);
        }

        // publish this step's row maxes into the other sPart buffer
#pragma unroll
        for (int r = 0; r < 8; ++r) {
            float v = alphaReg[r];
#pragma unroll
            for (int o = 1; o < 16; o <<= 1) v = fmaxf(v, __shfl_xor(v, o, 16));
            if (ln == 0) sPart[q ^ 1][lh * 8 + r][wave] = v;
        }

        // refill next 32-step mask chunk (other buffer)
        if (((t + 1) & 31) == 0 && (t + 1) < S_LEN && tid < 128) {
            int rr = tid >> 3, ri = (tid & 7) * 4;
            v4f mv = *(const v4f*)(mask + (size_t)(b0 + rr) * S_LEN + (t + 1) + ri);
            *(v4f*)&sMaskC[((t + 1) >> 5) & 1][rr][ri] = mv;
        }

        if (t + 2 < S_LEN) wait_async2();   // t+1's loads done; t+2's pending
        else               wait_async0();   // drain tail
        __syncthreads();                    // single per-step barrier
    }

    // ---- final logsumexp_j( alpha[b,j] + trans[j,END] ) ----
#pragma unroll
    for (int r = 0; r < 8; ++r)
        sExpA[0][lh * 8 + r][n0 + ln] = alphaReg[r];    // raw alpha now
    __syncthreads();
    if (tid < 16) {
        float mx = NEGV;
        for (int j = 0; j < T_TAGS; ++j)
            mx = fmaxf(mx, sExpA[0][tid][j] + trans[j * T_DIM + END_TAG]);
        float s = 0.0f;
        for (int j = 0; j < T_TAGS; ++j)
            s += __expf(sExpA[0][tid][j] + trans[j * T_DIM + END_TAG] - mx);
        log_den[b0 + tid] = mx + __logf(s);
    }
}

// ---------------------------------------------------------------------------
// Kernel 3: mean over batch of (log_den - log_num)
// ---------------------------------------------------------------------------
__global__ void __launch_bounds__(128)
crf_final(const float* __restrict__ log_den, const float* __restrict__ log_num,
          float* __restrict__ out) {
    __shared__ float sred[128];
    int tid = threadIdx.x;
    sred[tid] = log_den[tid] - log_num[tid];
    __syncthreads();
    for (int s = 64; s > 0; s >>= 1) {
        if (tid < s) sred[tid] += sred[tid + s];
        __syncthreads();
    }
    if (tid == 0) out[0] = sred[0] * (1.0f / (float)B_TOT);
}

// ---------------------------------------------------------------------------
extern "C" void kernel_launch(void* const* d_in, const int* in_sizes, int n_in,
                              void* d_out, int out_size, void* d_ws, size_t ws_size,
                              hipStream_t stream) {
    const float* emissions = (const float*)d_in[0];   // (128, 2048, 96) f32
    const int*   tags      = (const int*)  d_in[1];   // (128, 2048) int
    const float* mask      = (const float*)d_in[2];   // (128, 2048) f32
    const float* trans     = (const float*)d_in[3];   // (98, 98) f32

    float* ws      = (float*)d_ws;
    float* expT    = ws;                              // 96*96
    float* log_den = expT + T_TAGS * T_TAGS;          // 128
    float* log_num = log_den + B_TOT;                 // 128

    crf_exptrans<<<(T_TAGS * T_TAGS + 255) / 256, 256, 0, stream>>>(trans, expT);
    crf_score<<<B_TOT, 256, 0, stream>>>(emissions, tags, mask, trans, log_num);
    crf_partition<<<B_TOT / 16, 192, 0, stream>>>(emissions, mask, trans, expT, log_den);
    crf_final<<<1, 128, 0, stream>>>(log_den, log_num, (float*)d_out);
}